// GNN_GAT_72816875536864
// MI455X (gfx1250) — compile-verified
//
#include <hip/hip_runtime.h>
#include <hip/hip_bf16.h>
#include <math.h>

typedef __attribute__((ext_vector_type(2))) float v2f;
typedef __attribute__((ext_vector_type(8))) float v8f;

#define N_NODES 10000
#define N_EDGES 200000
#define HC 320     // HEADS * CONV
#define NHEADS 5
#define CONV 64
#define NEG_SLOPE 0.2f

// ---------------------------------------------------------------------------
// Self-loop attr: deg[d] += 1 ; loop_attr[d] += edge_attr[e]   (then divide)
// ---------------------------------------------------------------------------
__global__ void deg_sum_kernel(const int* __restrict__ dst,
                               const float* __restrict__ edge_attr,
                               float* __restrict__ deg,
                               float* __restrict__ loop_attr) {
    int e = blockIdx.x * blockDim.x + threadIdx.x;
    if (e >= N_EDGES) return;
    int d = dst[e];
    atomicAdd(&deg[d], 1.0f);
    const float* ea = edge_attr + (size_t)e * 16;
    float* la = loop_attr + (size_t)d * 16;
#pragma unroll
    for (int k = 0; k < 16; ++k) atomicAdd(&la[k], ea[k]);
}

__global__ void deg_norm_kernel(const float* __restrict__ deg,
                                float* __restrict__ loop_attr) {
    int t = blockIdx.x * blockDim.x + threadIdx.x;
    if (t >= N_NODES * 16) return;
    float inv = 1.0f / fmaxf(deg[t >> 4], 1.0f);
    loop_attr[t] *= inv;
}

// ---------------------------------------------------------------------------
// xl = A[M,K] @ W[K,320] + bias   via V_WMMA_F32_16X16X4_F32
// One wave computes one 16x16 output tile. M=10000 -> 625 tiles, N -> 20 tiles.
// A-operand: lane l -> row m0+(l&15), a[v] = A[row][k + v + 2*(l>>4)]
// B-operand: lane l -> col n0+(l&15), b[v] = W[k + v + 2*(l>>4)][col]
// D: vgpr r, lanes<16 -> row m0+r ; lanes>=16 -> row m0+8+r ; col n0+(l&15)
// ---------------------------------------------------------------------------
__global__ void gemm_xl_kernel(const float* __restrict__ A,
                               const float* __restrict__ W,
                               const float* __restrict__ bias,
                               float* __restrict__ out,
                               int M, int K) {
    int wave = (blockIdx.x * blockDim.x + threadIdx.x) >> 5;
    int lane = threadIdx.x & 31;
    const int NT = HC / 16;                    // 20 tiles along N
    int totalTiles = (M / 16) * NT;
    if (wave >= totalTiles) return;
    int mt = wave / NT, nt = wave % NT;
    int m0 = mt * 16, n0 = nt * 16;
    int half = lane >> 4;                      // 0 or 1
    int lidx = lane & 15;

    const float* arow = A + (size_t)(m0 + lidx) * K;
    const float* bcol = W + n0 + lidx;

    v8f c = {};
    for (int k = 0; k < K; k += 4) {
        int ka = k + 2 * half;
        v2f a, b;
        a.x = arow[ka];
        a.y = arow[ka + 1];
        b.x = bcol[(size_t)ka * HC];
        b.y = bcol[(size_t)(ka + 1) * HC];
        c = __builtin_amdgcn_wmma_f32_16x16x4_f32(false, a, false, b,
                                                  (short)0, c, false, false);
    }
    float bv = bias[n0 + lidx];
#pragma unroll
    for (int r = 0; r < 8; ++r) {
        int m = m0 + r + 8 * half;
        out[(size_t)m * HC + n0 + lidx] = c[r] + bv;
    }
}

// ---------------------------------------------------------------------------
// Per-edge attention score (fused edge embedding, K=16), wave per edge.
// score[e,h] = sum_c lrelu(xl[s,c]+xl[d,c]+ (ea@We)[c]) * att[h,c]
// Also: order-preserving uint-key atomicMax for per-destination max.
// ---------------------------------------------------------------------------
__device__ __forceinline__ unsigned f2key(float f) {
    unsigned u = __float_as_uint(f);
    return (u & 0x80000000u) ? ~u : (u | 0x80000000u);
}
__device__ __forceinline__ float key2f(unsigned k) {
    unsigned u = (k & 0x80000000u) ? (k & 0x7fffffffu) : ~k;
    return __uint_as_float(u);
}

__global__ void score_kernel(const int* __restrict__ src,
                             const int* __restrict__ dst,
                             const float* __restrict__ edge_attr,
                             const float* __restrict__ loop_attr,
                             const float* __restrict__ xl,
                             const float* __restrict__ We,   // [16,320]
                             const float* __restrict__ att,  // [5,64]
                             float* __restrict__ score,      // [E+N,5]
                             unsigned* __restrict__ smaxKey) // [N,5]
{
    int wid = (blockIdx.x * blockDim.x + threadIdx.x) >> 5;
    int lane = threadIdx.x & 31;
    const int Et = N_EDGES + N_NODES;
    if (wid >= Et) return;

    int s, d;
    const float* ea;
    if (wid < N_EDGES) {
        s = src[wid]; d = dst[wid];
        ea = edge_attr + (size_t)wid * 16;
    } else {
        s = d = wid - N_EDGES;
        ea = loop_attr + (size_t)(wid - N_EDGES) * 16;
    }
    float e[16];
#pragma unroll
    for (int k = 0; k < 16; ++k) e[k] = ea[k];

    const float* xs = xl + (size_t)s * HC;
    const float* xd = xl + (size_t)d * HC;

    for (int h = 0; h < NHEADS; ++h) {
        float acc = 0.f;
#pragma unroll
        for (int i = 0; i < 2; ++i) {
            int cc = i * 32 + lane;           // channel within head
            int c  = h * CONV + cc;           // absolute channel
            float emb = 0.f;
#pragma unroll
            for (int k = 0; k < 16; ++k) emb += e[k] * We[k * HC + c];
            float m = xs[c] + xd[c] + emb;
            m = (m > 0.f) ? m : NEG_SLOPE * m;
            acc += m * att[h * CONV + cc];
        }
#pragma unroll
        for (int off = 16; off > 0; off >>= 1) acc += __shfl_xor(acc, off, 32);
        if (lane == 0) {
            score[(size_t)wid * NHEADS + h] = acc;
            atomicMax(&smaxKey[d * NHEADS + h], f2key(acc));
        }
    }
}

// ---------------------------------------------------------------------------
// ex = exp(score - smax[dst]); denom[dst,h] += ex  (score overwritten by ex)
// ---------------------------------------------------------------------------
__global__ void expdenom_kernel(const int* __restrict__ dst,
                                float* __restrict__ score,
                                const unsigned* __restrict__ smaxKey,
                                float* __restrict__ denom) {
    int t = blockIdx.x * blockDim.x + threadIdx.x;
    const int Et = N_EDGES + N_NODES;
    if (t >= Et * NHEADS) return;
    int e = t / NHEADS, h = t - e * NHEADS;
    int d = (e < N_EDGES) ? dst[e] : (e - N_EDGES);
    float smax = key2f(smaxKey[d * NHEADS + h]);
    float ex = expf(score[t] - smax);
    score[t] = ex;
    atomicAdd(&denom[d * NHEADS + h], ex);
}

// ---------------------------------------------------------------------------
// acc[d, c] += (ex/denom) * xl[s, c]   — wave per edge, 10 channels per lane
// ---------------------------------------------------------------------------
__global__ void aggregate_kernel(const int* __restrict__ src,
                                 const int* __restrict__ dst,
                                 const float* __restrict__ ex,    // [E+N,5]
                                 const float* __restrict__ denom, // [N,5]
                                 const float* __restrict__ xl,
                                 float* __restrict__ acc) {
    int wid = (blockIdx.x * blockDim.x + threadIdx.x) >> 5;
    int lane = threadIdx.x & 31;
    const int Et = N_EDGES + N_NODES;
    if (wid >= Et) return;
    int s, d;
    if (wid < N_EDGES) { s = src[wid]; d = dst[wid]; }
    else               { s = d = wid - N_EDGES; }
    float alpha[NHEADS];
#pragma unroll
    for (int h = 0; h < NHEADS; ++h)
        alpha[h] = ex[(size_t)wid * NHEADS + h] / denom[d * NHEADS + h];
    const float* xs = xl + (size_t)s * HC;
    float* od = acc + (size_t)d * HC;
#pragma unroll
    for (int i = 0; i < 10; ++i) {
        int c = i * 32 + lane;
        atomicAdd(&od[c], alpha[c >> 6] * xs[c]);
    }
}

// ---------------------------------------------------------------------------
// out[n,c] = elu( mean_h acc[n,h,c] + bias[c] )
// ---------------------------------------------------------------------------
__global__ void finalize_kernel(const float* __restrict__ acc,
                                const float* __restrict__ bias,
                                float* __restrict__ out) {
    int t = blockIdx.x * blockDim.x + threadIdx.x;
    if (t >= N_NODES * CONV) return;
    int n = t / CONV, c = t - n * CONV;
    const float* a = acc + (size_t)n * HC + c;
    float v = (a[0] + a[64] + a[128] + a[192] + a[256]) * 0.2f + bias[c];
    out[t] = (v > 0.f) ? v : expm1f(v);
}

// ---------------------------------------------------------------------------
extern "C" void kernel_launch(void* const* d_in, const int* in_sizes, int n_in,
                              void* d_out, int out_size, void* d_ws, size_t ws_size,
                              hipStream_t stream) {
    const float* x         = (const float*)d_in[0];
    const int*   eidx      = (const int*)d_in[1];
    const float* edge_attr = (const float*)d_in[2];
    const int* src = eidx;
    const int* dst = eidx + N_EDGES;

    // workspace carving (256B aligned)
    char* p = (char*)d_ws;
    auto alloc = [&](size_t bytes) -> void* {
        void* r = (void*)p;
        p += (bytes + 255) & ~(size_t)255;
        return r;
    };
    float*    xl        = (float*)alloc((size_t)N_NODES * HC * 4);
    float*    acc       = (float*)alloc((size_t)N_NODES * HC * 4);
    float*    score     = (float*)alloc((size_t)(N_EDGES + N_NODES) * NHEADS * 4);
    unsigned* smaxKey   = (unsigned*)alloc((size_t)N_NODES * NHEADS * 4);
    float*    denom     = (float*)alloc((size_t)N_NODES * NHEADS * 4);
    float*    deg       = (float*)alloc((size_t)N_NODES * 4);
    float*    loop_attr = (float*)alloc((size_t)N_NODES * 16 * 4);
    float*    h1        = (float*)alloc((size_t)N_NODES * CONV * 4);
    float*    h2        = (float*)alloc((size_t)N_NODES * CONV * 4);

    const int Et = N_EDGES + N_NODES;

    // ---- self-loop attr (layer independent) ----
    hipMemsetAsync(deg, 0, (size_t)N_NODES * 4, stream);
    hipMemsetAsync(loop_attr, 0, (size_t)N_NODES * 16 * 4, stream);
    deg_sum_kernel<<<(N_EDGES + 255) / 256, 256, 0, stream>>>(dst, edge_attr, deg, loop_attr);
    deg_norm_kernel<<<(N_NODES * 16 + 255) / 256, 256, 0, stream>>>(deg, loop_attr);

    const float* hin = x;
    int K = 128;
    float* houts[3] = {h1, h2, (float*)d_out};

    for (int l = 0; l < 3; ++l) {
        const float* Wl   = (const float*)d_in[3 + 5 * l];
        const float* bl   = (const float*)d_in[4 + 5 * l];
        const float* We   = (const float*)d_in[5 + 5 * l];
        const float* att  = (const float*)d_in[6 + 5 * l];
        const float* bias = (const float*)d_in[7 + 5 * l];

        hipMemsetAsync(smaxKey, 0, (size_t)N_NODES * NHEADS * 4, stream); // key 0 == -NaN (minimum)
        hipMemsetAsync(denom, 0, (size_t)N_NODES * NHEADS * 4, stream);
        hipMemsetAsync(acc, 0, (size_t)N_NODES * HC * 4, stream);

        // node GEMM: one 16x16 tile per wave, 8 waves/block
        int tiles = (N_NODES / 16) * (HC / 16);   // 625*20 = 12500
        gemm_xl_kernel<<<(tiles + 7) / 8, 256, 0, stream>>>(hin, Wl, bl, xl, N_NODES, K);

        // attention score + segment max  (wave per edge, 8 waves/block)
        score_kernel<<<(Et + 7) / 8, 256, 0, stream>>>(src, dst, edge_attr, loop_attr,
                                                       xl, We, att, score, smaxKey);
        // softmax numerator + denominator
        expdenom_kernel<<<(Et * NHEADS + 255) / 256, 256, 0, stream>>>(dst, score, smaxKey, denom);
        // weighted scatter-add
        aggregate_kernel<<<(Et + 7) / 8, 256, 0, stream>>>(src, dst, score, denom, xl, acc);
        // head mean + bias + ELU
        finalize_kernel<<<(N_NODES * CONV + 255) / 256, 256, 0, stream>>>(acc, bias, houts[l]);

        hin = houts[l];
        K = CONV;
    }
}